// GNNLayer_46514495816146
// MI455X (gfx1250) — compile-verified
//
#include <hip/hip_runtime.h>
#include <hip/hip_bf16.h>

typedef __attribute__((ext_vector_type(2))) float v2f;
typedef __attribute__((ext_vector_type(4))) float v4f;
typedef __attribute__((ext_vector_type(8))) float v8f;

#define D 128           // D_IN == D_OUT == 128
#define LN_EPS 1e-5f

// ---------------------------------------------------------------- degree init
__global__ void __launch_bounds__(256)
deg_init_kernel(float* __restrict__ deg, int N) {
    int i = blockIdx.x * blockDim.x + threadIdx.x;
    if (i < N) deg[i] = 1.0f;   // self-loop weight
}

// ------------------------------------------------------------- degree scatter
__global__ void __launch_bounds__(256)
deg_scatter_kernel(const int* __restrict__ cols, const float* __restrict__ ew,
                   float* __restrict__ deg, int E) {
    int e = blockIdx.x * blockDim.x + threadIdx.x;
    if (e < E) atomicAdd(&deg[cols[e]], ew[e]);
}

// -------------------------------------------------------------- deg -> rsqrt
__global__ void __launch_bounds__(256)
dinv_kernel(float* __restrict__ deg, int N) {
    int i = blockIdx.x * blockDim.x + threadIdx.x;
    if (i < N) {
        float d = deg[i];
        deg[i] = (d > 0.0f) ? rsqrtf(d) : 0.0f;
    }
}

// ------------------------------------------------- h = x @ W^T  (f32 WMMA)
// One wave per 16x16 tile. A = x[16x4], B = W^T[4x16] (B[k][n] = W[n][k]).
// ISA frag layout (32-bit 16x4 A): lanes 0-15 hold M=lane, VGPR0/1 = K 0/1;
// lanes 16-31 hold K 2/3. B mirrors this with N=lane&15. C/D: VGPR r holds
// M=r (lanes 0-15) / M=r+8 (lanes 16-31), N=lane&15.
__global__ void __launch_bounds__(256)
gemm_wmma_kernel(const float* __restrict__ x, const float* __restrict__ W,
                 const float* __restrict__ dinv, float* __restrict__ h,
                 float* __restrict__ acc, int N) {
    const int lane  = threadIdx.x;            // 0..31
    const int nbase = threadIdx.y * 16;       // 8 waves cover all 128 cols
    const int mbase = blockIdx.x * 16;
    const int lmod  = lane & 15;
    const int koff  = (lane >> 4) * 2;        // K 0/1 vs K 2/3

    // clamp load row so the tail wave stays fully active (EXEC must be ~0 for WMMA)
    int arow = mbase + lmod;
    if (arow > N - 1) arow = N - 1;

    const float* xrow = x + (size_t)arow * D + koff;
    const float* wrow = W + (size_t)(nbase + lmod) * D + koff;

    v8f c = {};
#pragma unroll
    for (int k0 = 0; k0 < D; k0 += 4) {
        v2f a = *(const v2f*)(xrow + k0);
        v2f b = *(const v2f*)(wrow + k0);
        // 8 args: (neg_a, A, neg_b, B, c_mod, C, reuse_a, reuse_b)
        c = __builtin_amdgcn_wmma_f32_16x16x4_f32(false, a, false, b,
                                                  (short)0, c, false, false);
    }

    const int mhi = (lane >> 4) * 8;
#pragma unroll
    for (int r = 0; r < 8; ++r) {
        int row = mbase + mhi + r;
        if (row < N) {
            int col = nbase + lmod;
            float v  = c[r];
            float di = dinv[row];
            h[(size_t)row * D + col]   = v;
            acc[(size_t)row * D + col] = v * di * di;  // self-loop contribution
        }
    }
}

// ------------------------------------------------- edge gather + scatter-add
// One wave per edge; lane l owns channels [4l, 4l+4): 512B contiguous per wave.
__global__ void __launch_bounds__(256)
edge_scatter_kernel(const int* __restrict__ rows, const int* __restrict__ cols,
                    const float* __restrict__ ew, const float* __restrict__ dinv,
                    const float* __restrict__ h, float* __restrict__ acc, int E) {
    long long t = (long long)blockIdx.x * blockDim.x + threadIdx.x;
    int e = (int)(t >> 5);
    if (e >= E) return;
    int part = (int)(t & 31);

    int   src  = rows[e];
    int   dst  = cols[e];
    float norm = dinv[src] * ew[e] * dinv[dst];

    v4f hv = *(const v4f*)(h + (size_t)src * D + part * 4);
    float* out = acc + (size_t)dst * D + part * 4;
#pragma unroll
    for (int i = 0; i < 4; ++i)
        atomicAdd(out + i, hv[i] * norm);
}

// ------------------------------------- bias + ReLU + LayerNorm (wave32/node)
__global__ void __launch_bounds__(256)
finalize_kernel(float* __restrict__ out, const float* __restrict__ bias,
                const float* __restrict__ gamma, const float* __restrict__ beta,
                int N) {
    int lane = threadIdx.x & 31;
    int wave = threadIdx.x >> 5;
    int node = blockIdx.x * 8 + wave;
    if (node >= N) return;

    float* p = out + (size_t)node * D + lane * 4;
    v4f v = *(const v4f*)p;
    float s = 0.f, s2 = 0.f;
#pragma unroll
    for (int i = 0; i < 4; ++i) {
        int c = lane * 4 + i;
        float vv = v[i] + bias[c];
        vv = fmaxf(vv, 0.0f);
        v[i] = vv;
        s  += vv;
        s2 += vv * vv;
    }
#pragma unroll
    for (int off = 16; off >= 1; off >>= 1) {
        s  += __shfl_xor(s,  off, 32);
        s2 += __shfl_xor(s2, off, 32);
    }
    float mu  = s * (1.0f / 128.0f);
    float var = s2 * (1.0f / 128.0f) - mu * mu;
    float rs  = rsqrtf(var + LN_EPS);
#pragma unroll
    for (int i = 0; i < 4; ++i) {
        int c = lane * 4 + i;
        v[i] = (v[i] - mu) * rs * gamma[c] + beta[c];
    }
    *(v4f*)p = v;
}

// ---------------------------------------------------------------------------
extern "C" void kernel_launch(void* const* d_in, const int* in_sizes, int n_in,
                              void* d_out, int out_size, void* d_ws, size_t ws_size,
                              hipStream_t stream) {
    const float* x     = (const float*)d_in[0];
    const int*   eidx  = (const int*)  d_in[1];   // [2, E] int32 (JAX x64 off)
    const float* ew    = (const float*)d_in[2];
    const float* W     = (const float*)d_in[3];
    const float* bias  = (const float*)d_in[4];
    const float* gamma = (const float*)d_in[5];
    const float* beta  = (const float*)d_in[6];
    float*       out   = (float*)d_out;

    const int N = in_sizes[0] / D;
    const int E = in_sizes[2];
    const int* rows = eidx;       // sources
    const int* cols = eidx + E;   // targets

    float* h    = (float*)d_ws;                 // N*128 f32
    float* dinv = h + (size_t)N * D;            // N f32 (deg, then rsqrt in place)

    deg_init_kernel   <<<(N + 255) / 256, 256, 0, stream>>>(dinv, N);
    deg_scatter_kernel<<<(E + 255) / 256, 256, 0, stream>>>(cols, ew, dinv, E);
    dinv_kernel       <<<(N + 255) / 256, 256, 0, stream>>>(dinv, N);

    gemm_wmma_kernel<<<(N + 15) / 16, dim3(32, 8), 0, stream>>>(x, W, dinv, h, out, N);

    long long sthreads = (long long)E * 32;
    edge_scatter_kernel<<<(unsigned)((sthreads + 255) / 256), 256, 0, stream>>>(
        rows, cols, ew, dinv, h, out, E);

    finalize_kernel<<<(N + 7) / 8, 256, 0, stream>>>(out, bias, gamma, beta, N);
}